// SatoFilter_51762945851825
// MI455X (gfx1250) — compile-verified
//
#include <hip/hip_runtime.h>
#include <hip/hip_bf16.h>
#include <math.h>

// ---------------------------------------------------------------------------
// Sato ridge filter (5 sigmas) built on CDNA5 V_WMMA_F32_16X16X4_F32.
// 1-D separable convolutions are expressed as banded-Toeplitz matmuls:
//   out[16 x 16] += A_toeplitz[16 x 4] * B_tile[4 x 16]   (accumulated over K)
// Vertical pass writes results transposed (LDS transpose) so the horizontal
// pass is again a "vertical" matmul with coalesced loads.
// ---------------------------------------------------------------------------

typedef float v2f __attribute__((ext_vector_type(2)));
typedef float v8f __attribute__((ext_vector_type(8)));

#define H_IMG   2048
#define W_IMG   2048
#define RMAX    36
#define GPW     2128            // padded gray width (stride), 133*16
#define GPH     2120            // padded gray height (2048 + 2*36)
#define GTS     2048            // Gt row stride (y dimension)
#define NPIX    (2048*2048)

__device__ __forceinline__ v8f wmma4(v2f a, v2f b, v8f c) {
    // D = A(16x4, f32) * B(4x16, f32) + C(16x16, f32)
    return __builtin_amdgcn_wmma_f32_16x16x4_f32(
        false, a, false, b, (short)0, c, false, false);
}

// ---------------------------------------------------------------------------
// Build Gaussian-derivative kernels (double, like scipy) and pack the
// 16 x (L+15) Toeplitz A matrices in WMMA A-operand lane order:
//   element (chunk kc, lane, v) -> K = 4*kc + v + 2*(lane>>4), M = lane&15,
//   value = w[K-M] if 0 <= K-M < L else 0.
// Layout: apk + P[s]*3*64  then [ord][chunk][lane][v].
// ---------------------------------------------------------------------------
__global__ void init_apk_kernel(float* __restrict__ apk) {
    int s = threadIdx.x;
    if (s >= 5) return;
    const int sigma = 2 * s + 1;
    const int r = 4 * sigma;          // int(4*sigma + 0.5) for odd sigma
    const int L = 2 * r + 1;          // <= 73
    const double s2 = (double)sigma * (double)sigma;

    double phi[73];
    double sum = 0.0;
    for (int t = 0; t < L; ++t) {
        double x = (double)(t - r);
        phi[t] = exp(-0.5 * x * x / s2);
        sum += phi[t];
    }
    float w[3][73];
    for (int t = 0; t < L; ++t) {
        double p = phi[t] / sum;
        double x = (double)(t - r);
        double k0 = p;
        double k1 = -x / s2 * p;
        double k2 = (x * x / (s2 * s2) - 1.0 / s2) * p;
        // reversed (scipy correlate1d(weights=kernel[::-1]))
        w[0][L - 1 - t] = (float)k0;
        w[1][L - 1 - t] = (float)k1;
        w[2][L - 1 - t] = (float)k2;
    }
    const int chunks = 2 * sigma + 4;          // (L+15)/4
    const int P[5] = {0, 6, 16, 30, 48};       // prefix of chunks
    float* base = apk + (size_t)P[s] * 3 * 64;
    for (int ord = 0; ord < 3; ++ord)
        for (int c = 0; c < chunks; ++c)
            for (int lane = 0; lane < 32; ++lane)
                for (int v = 0; v < 2; ++v) {
                    int K = 4 * c + v + 2 * (lane >> 4);
                    int M = lane & 15;
                    int j = K - M;
                    float val = (j >= 0 && j < L) ? w[ord][j] : 0.0f;
                    base[(((size_t)ord * chunks + c) * 32 + lane) * 2 + v] = val;
                }
}

// Grayscale + symmetric (reflect) pad into GPH x GPW plane.
__global__ void __launch_bounds__(256) gray_pad_kernel(
    const float* __restrict__ img, float* __restrict__ gp) {
    unsigned idx = blockIdx.x * 256u + threadIdx.x;
    const unsigned total = (unsigned)GPH * GPW;
    if (idx >= total) return;
    int xp = (int)(idx % GPW);
    int yp = (int)(idx / GPW);
    if (xp >= 2048 + 2 * RMAX) { gp[idx] = 0.0f; return; }
    int y = yp - RMAX; y = (y < 0) ? (-y - 1) : ((y >= H_IMG) ? (2 * H_IMG - 1 - y) : y);
    int x = xp - RMAX; x = (x < 0) ? (-x - 1) : ((x >= W_IMG) ? (2 * W_IMG - 1 - x) : x);
    int p = y * W_IMG + x;
    gp[idx] = 0.2989f * img[p] + 0.587f * img[NPIX + p] + 0.114f * img[2 * NPIX + p];
}

// ---------------------------------------------------------------------------
// Pass 1 (vertical conv): for each padded column tile, compute three
// convolutions (k2, k1, k0) of the padded gray image sharing one B stream.
// Outputs stored TRANSPOSED: Gt[xp][y], stride GTS.
// Grid: 2128 blocks x 256 threads = 17024 waves = 133(x) * 128(y) tiles.
// ---------------------------------------------------------------------------
__global__ void __launch_bounds__(256) sato_pass1_kernel(
    const float* __restrict__ gp,
    float* __restrict__ g0, float* __restrict__ g1, float* __restrict__ g2,
    const float* __restrict__ a2, const float* __restrict__ a1,
    const float* __restrict__ a0, int chunks, int r) {
    __shared__ float T[3][8][16][17];
    const int lane = threadIdx.x & 31;
    const int wave = threadIdx.x >> 5;
    const int m = lane & 15;
    const int h = lane >> 4;
    const int t = blockIdx.x * 8 + wave;
    const int xt = t % 133;
    const int yt = t / 133;
    const int x0 = xt * 16, y0 = yt * 16;
    const int base = y0 + RMAX - r;

    const float* bp = gp + (size_t)(base + 2 * h) * GPW + x0 + m;
    v8f c0 = {0,0,0,0,0,0,0,0}, c1 = c0, c2 = c0;

    for (int kc = 0; kc < chunks; ++kc) {
        v2f b;
        b.x = bp[0];
        b.y = bp[GPW];
        const int ai = (kc * 32 + lane) * 2;
        v2f A2 = *(const v2f*)(a2 + ai);
        v2f A1 = *(const v2f*)(a1 + ai);
        v2f A0 = *(const v2f*)(a0 + ai);
        c0 = wmma4(A2, b, c0);   // order-2 -> g0
        c1 = wmma4(A1, b, c1);   // order-1 -> g1
        c2 = wmma4(A0, b, c2);   // order-0 -> g2
        bp += 4 * GPW;
    }
    // C element (v, lane) = out[M = v + 8h][N = m]; store T[M][N].
    for (int v = 0; v < 8; ++v) {
        T[0][wave][v + 8 * h][m] = c0[v];
        T[1][wave][v + 8 * h][m] = c1[v];
        T[2][wave][v + 8 * h][m] = c2[v];
    }
    __syncthreads();
    // Gt[x0+row][y0+m] = out[y_off = m][x_off = row] = T[m][row]
    for (int rr = 0; rr < 8; ++rr) {
        int row = 2 * rr + h;
        int gi = (x0 + row) * GTS + y0 + m;
        g0[gi] = T[0][wave][m][row];
        g1[gi] = T[1][wave][m][row];
        g2[gi] = T[2][wave][m][row];
    }
}

// ---------------------------------------------------------------------------
// Pass 2 (horizontal conv on transposed planes) + eigenvalue epilogue +
// running max over sigmas. Output in natural [y][x] layout.
// Grid: 2048 blocks x 256 threads = 16384 waves = 128 * 128 tiles.
// ---------------------------------------------------------------------------
__global__ void __launch_bounds__(256) sato_pass2_kernel(
    const float* __restrict__ g0, const float* __restrict__ g1,
    const float* __restrict__ g2, float* __restrict__ F,
    const float* __restrict__ a0, const float* __restrict__ a1,
    const float* __restrict__ a2, int chunks, int r, float sig2, int first) {
    __shared__ float T[8][16][17];
    const int lane = threadIdx.x & 31;
    const int wave = threadIdx.x >> 5;
    const int m = lane & 15;
    const int h = lane >> 4;
    const int t = blockIdx.x * 8 + wave;
    const int xt = t % 128;
    const int yt = t / 128;
    const int x0 = xt * 16, y0 = yt * 16;
    const int base = x0 + RMAX - r;

    const float* b0p = g0 + (size_t)(base + 2 * h) * GTS + y0 + m;
    const float* b1p = g1 + (size_t)(base + 2 * h) * GTS + y0 + m;
    const float* b2p = g2 + (size_t)(base + 2 * h) * GTS + y0 + m;
    v8f c0 = {0,0,0,0,0,0,0,0}, c1 = c0, c2 = c0;

    for (int kc = 0; kc < chunks; ++kc) {
        const int ai = (kc * 32 + lane) * 2;
        v2f A0 = *(const v2f*)(a0 + ai);
        v2f A1 = *(const v2f*)(a1 + ai);
        v2f A2 = *(const v2f*)(a2 + ai);
        v2f B0; B0.x = b0p[0]; B0.y = b0p[GTS];
        v2f B1; B1.x = b1p[0]; B1.y = b1p[GTS];
        v2f B2; B2.x = b2p[0]; B2.y = b2p[GTS];
        c0 = wmma4(A0, B0, c0);  // hrr = conv(G0, k0)
        c1 = wmma4(A1, B1, c1);  // hrc = conv(G1, k1)
        c2 = wmma4(A2, B2, c2);  // hcc = conv(G2, k2)
        b0p += 4 * GTS; b1p += 4 * GTS; b2p += 4 * GTS;
    }
    for (int v = 0; v < 8; ++v) {
        float hrr = c0[v], hrc = c1[v], hcc = c2[v];
        float mid = 0.5f * (hrr + hcc);
        float dif = 0.5f * (hrr - hcc);
        float s = sqrtf(dif * dif + hrc * hrc);
        float lam = mid - s;
        float val = sig2 * fmaxf(-lam, 0.0f);
        T[wave][v + 8 * h][m] = val;   // T[M = x_off][N = y_off]
    }
    __syncthreads();
    for (int rr = 0; rr < 8; ++rr) {
        int row = 2 * rr + h;                         // y_off
        int idx = (y0 + row) * W_IMG + x0 + m;        // natural layout
        float nv = T[wave][m][row];                   // (x_off=m, y_off=row)
        float ov = first ? 0.0f : F[idx];
        F[idx] = fmaxf(ov, nv);
    }
}

__global__ void init_minmax_kernel(unsigned* mm) {
    mm[0] = 0x7f800000u;  // +inf  (values are >= 0, uint order == float order)
    mm[1] = 0u;
}

__global__ void __launch_bounds__(256) reduce_minmax_kernel(
    const float* __restrict__ F, unsigned* __restrict__ mm) {
    __shared__ float smn[256], smx[256];
    unsigned tid = threadIdx.x;
    unsigned i = blockIdx.x * 256u + tid;
    unsigned stride = gridDim.x * 256u;
    float mn = 3.402823e38f, mx = 0.0f;
    for (; i < (unsigned)NPIX; i += stride) {
        float v = F[i];
        mn = fminf(mn, v);
        mx = fmaxf(mx, v);
    }
    smn[tid] = mn; smx[tid] = mx;
    __syncthreads();
    for (int s = 128; s > 0; s >>= 1) {
        if ((int)tid < s) {
            smn[tid] = fminf(smn[tid], smn[tid + s]);
            smx[tid] = fmaxf(smx[tid], smx[tid + s]);
        }
        __syncthreads();
    }
    if (tid == 0) {
        atomicMin(&mm[0], __float_as_uint(smn[0]));
        atomicMax(&mm[1], __float_as_uint(smx[0]));
    }
}

__global__ void __launch_bounds__(256) normalize_kernel(
    const float* __restrict__ F, const unsigned* __restrict__ mm,
    float* __restrict__ out) {
    unsigned i = blockIdx.x * 256u + threadIdx.x;
    if (i >= (unsigned)NPIX) return;
    float mn = __uint_as_float(mm[0]);
    float mx = __uint_as_float(mm[1]);
    out[i] = (F[i] - mn) / (mx - mn);
}

// ---------------------------------------------------------------------------
extern "C" void kernel_launch(void* const* d_in, const int* in_sizes, int n_in,
                              void* d_out, int out_size, void* d_ws, size_t ws_size,
                              hipStream_t stream) {
    const float* img = (const float*)d_in[0];
    float* out = (float*)d_out;
    float* ws = (float*)d_ws;

    // Workspace layout (floats):
    const size_t OFF_APK  = 64;                      // Toeplitz packs (13440 f)
    const size_t OFF_GRAY = 16384;                   // padded gray
    const size_t GRAY_SZ  = (size_t)GPH * GPW;       // 4,511,360
    const size_t PLANE_G  = (size_t)GPW * GTS;       // 4,358,144
    unsigned* mm = (unsigned*)ws;
    float* apk = ws + OFF_APK;
    float* gp  = ws + OFF_GRAY;
    float* g0  = gp + GRAY_SZ;
    float* g1  = g0 + PLANE_G;
    float* g2  = g1 + PLANE_G;
    float* F   = g2 + PLANE_G;

    init_apk_kernel<<<1, 32, 0, stream>>>(apk);
    gray_pad_kernel<<<(GPH * GPW + 255) / 256, 256, 0, stream>>>(img, gp);
    init_minmax_kernel<<<1, 1, 0, stream>>>(mm);

    const int P[5] = {0, 6, 16, 30, 48};
    for (int s = 0; s < 5; ++s) {
        int sigma = 2 * s + 1;
        int r = 4 * sigma;
        int chunks = 2 * sigma + 4;
        const float* base = apk + (size_t)P[s] * 3 * 64;
        const float* A0 = base;
        const float* A1 = base + (size_t)chunks * 64;
        const float* A2 = base + (size_t)2 * chunks * 64;
        sato_pass1_kernel<<<2128, 256, 0, stream>>>(gp, g0, g1, g2,
                                                    A2, A1, A0, chunks, r);
        sato_pass2_kernel<<<2048, 256, 0, stream>>>(g0, g1, g2, F,
                                                    A0, A1, A2, chunks, r,
                                                    (float)(sigma * sigma),
                                                    s == 0 ? 1 : 0);
    }
    reduce_minmax_kernel<<<1024, 256, 0, stream>>>(F, mm);
    normalize_kernel<<<(NPIX + 255) / 256, 256, 0, stream>>>(F, mm, out);
}